// PositionalEncoding_57672820851185
// MI455X (gfx1250) — compile-verified
//
#include <hip/hip_runtime.h>

#define MEM_LEN 30
#define CAN_LEN 200
#define NC      99            // can_bboxes per batch (NUM_CAND-1)
#define D       128
#define ROWS    (MEM_LEN + CAN_LEN)   // 230
#define PE_Y    211
#define PE_Z    61

// One wave32 per output row: gather the 256B fp16 PE row via the CDNA5
// async global->LDS path (ASYNCcnt), overlap with the 512B f32 input row
// load, then fuse convert+add and store b128. Pure HBM-roofline kernel.
__global__ __launch_bounds__(256) void pe_add_rows_kernel(
    const float* __restrict__ mem,
    const float* __restrict__ can,
    const float* __restrict__ mem_bb,
    const float* __restrict__ can_bb,
    const _Float16* __restrict__ pe,
    float* __restrict__ out,
    int total_rows)
{
    __shared__ __align__(16) unsigned char smem[8 * 256];   // 256B slot per wave

    const int lane = threadIdx.x & 31;
    const int wave = threadIdx.x >> 5;
    const int row  = blockIdx.x * 8 + wave;
    if (row >= total_rows) return;

    const int b   = row / ROWS;
    const int pos = row - b * ROWS;

    // reference bbox = mem_bboxes[b, MEM_LEN-1]
    const float4 ref = *(const float4*)(mem_bb + ((size_t)b * MEM_LEN + (MEM_LEN - 1)) * 4);

    float4 bb;
    int ti;
    const float* src;
    if (pos < MEM_LEN) {
        bb = *(const float4*)(mem_bb + ((size_t)b * MEM_LEN + pos) * 4);
        int tv = 2 * (pos - (MEM_LEN - 1));          // clip_idx(arange(-29..0), 2.0, 30)
        tv = tv < -30 ? -30 : (tv > 30 ? 30 : tv);
        ti = tv + 30;
        src = mem + ((size_t)b * MEM_LEN + pos) * D;
    } else {
        const int j = pos - MEM_LEN;                 // 0..199 in can section
        if (j < 2 * NC && (j & 1)) {
            bb = *(const float4*)(can_bb + ((size_t)b * NC + (j >> 1)) * 4);
        } else {
            bb = ref;                                // interleaved ref slots + 2 tail refs
        }
        ti = (j & 1) ? 34 : 32;                      // clip_idx(((j%2)+1)*2, 30) + 30
        src = can + ((size_t)b * CAN_LEN + j) * D;
    }

    // _distance_values in f32, matching the reference exactly
    const float w_r  = ref.z - ref.x + 1.0f;
    const float h_r  = ref.w - ref.y + 1.0f;
    const float cx_r = (ref.x + ref.z) * 0.5f;
    const float cy_r = (ref.y + ref.w) * 0.5f;
    const float w  = bb.z - bb.x + 1.0f;
    const float h  = bb.w - bb.y + 1.0f;
    const float cx = (bb.x + bb.z) * 0.5f;
    const float cy = (bb.y + bb.w) * 0.5f;
    const float dx = (cx - cx_r) / w_r;
    const float dy = (cy - cy_r) / h_r;
    const float xy = __builtin_sqrtf(dx * dx + dy * dy);
    const float sz = 0.5f * __builtin_logf((w * h) / (w_r * h_r));

    const int xi = (int)fminf(fmaxf(xy * 15.0f, -105.0f), 105.0f) + 105; // trunc-to-zero == astype(int32)
    const int si = (int)fminf(fmaxf(sz * 15.0f, -105.0f), 105.0f) + 105;

    const size_t pe_off = (((size_t)xi * PE_Y + (size_t)si) * PE_Z + (size_t)ti) * (size_t)D;

    // CDNA5 async copy: 8B/lane of the fp16 PE row -> this wave's LDS slot.
    // Generic->LDS-offset: low 32 bits of the flat pointer are the LDS byte
    // offset (aperture mapping, ISA 10.2). EXEC is all-1s here (wave-uniform row).
    unsigned int lds_off = (unsigned int)(size_t)(void*)(&smem[wave * 256 + lane * 8]);
    const void* gaddr = (const void*)((const char*)pe + pe_off * 2 + (size_t)lane * 8);
    asm volatile("global_load_async_to_lds_b64 %0, %1, off"
                 :: "v"(lds_off), "v"(gaddr) : "memory");

    // Overlap: 16B/lane of the f32 input row while the async copy is in flight.
    const float4 inv = *(const float4*)(src + lane * 4);

    asm volatile("s_wait_asynccnt 0" ::: "memory");

    union { unsigned long long u; _Float16 hv[4]; } pv;
    pv.u = *(const unsigned long long*)(&smem[wave * 256 + lane * 8]);

    float4 o;
    o.x = inv.x + (float)pv.hv[0];
    o.y = inv.y + (float)pv.hv[1];
    o.z = inv.z + (float)pv.hv[2];
    o.w = inv.w + (float)pv.hv[3];

    float* dst = out + (size_t)row * D;   // row == b*230 + pos, flat (B,230,128)
    *(float4*)(dst + lane * 4) = o;
}

extern "C" void kernel_launch(void* const* d_in, const int* in_sizes, int n_in,
                              void* d_out, int out_size, void* d_ws, size_t ws_size,
                              hipStream_t stream) {
    const float*    mem    = (const float*)d_in[0];
    const float*    can    = (const float*)d_in[1];
    const float*    mem_bb = (const float*)d_in[2];
    const float*    can_bb = (const float*)d_in[3];
    const _Float16* pe     = (const _Float16*)d_in[4];
    // d_in[5] = num_candidates (100) — structure is baked into the kernel.

    const int B          = in_sizes[0] / (MEM_LEN * D);
    const int total_rows = B * ROWS;
    const int blocks     = (total_rows + 7) / 8;      // 8 waves (rows) per 256-thread block

    pe_add_rows_kernel<<<blocks, 256, 0, stream>>>(
        mem, can, mem_bb, can_bb, pe, (float*)d_out, total_rows);
}